// LPDSSEmbedder_32693291057689
// MI455X (gfx1250) — compile-verified
//
#include <hip/hip_runtime.h>
#include <math.h>

// ---------------- problem constants (from reference) ----------------
#define FRAME 160
#define KFR   15
#define PCH   64            // P
#define ORDER 12
#define TLEN  (FRAME*KFR)   // 2400
#define LBLK  38            // ceil(T/P)
#define SSL   0.31622776601683794f   // 10^(-10/20)

typedef float v2f __attribute__((ext_vector_type(2)));
typedef float v8f __attribute__((ext_vector_type(8)));

__device__ __forceinline__ float wred32(float v) {
#pragma unroll
  for (int o = 16; o > 0; o >>= 1) v += __shfl_xor(v, o, 32);
  return v;
}

// =====================================================================
// Kernel 1: one wave32 per frame. Windowed frame in registers (5/lane),
// Burg recursion fully unrolled (shrinking windows constant-fold),
// reductions via shfl. Residual FIR via a per-wave LDS copy of the
// windowed frame. Per-frame sum/sumsq written for deterministic std.
// =====================================================================
__global__ __launch_bounds__(256)
void burg_resid_kernel(const float* __restrict__ pcm,
                       float* __restrict__ resid,
                       float* __restrict__ fsum,
                       float* __restrict__ fsq,
                       int nframes)
{
  const int wv   = threadIdx.x >> 5;
  const int lane = threadIdx.x & 31;
  const int fr   = blockIdx.x * 8 + wv;
  __shared__ float sx[8][176];     // 12 zero-pad + 160 windowed samples, per wave
  if (fr >= nframes) return;

  // ---- load + Hann window: lane holds samples 5*lane .. 5*lane+4 ----
  const float c2pi = 6.283185307179586f / 159.0f;
  float xw[5];
#pragma unroll
  for (int q = 0; q < 5; q++) {
    int i = lane * 5 + q;
    float w = 0.5f - 0.5f * cosf(c2pi * (float)i);
    xw[q] = pcm[(size_t)fr * FRAME + i] * w;
  }

  // ---- init fwd = x[1:], bwd = x[:-1]; invalid tail lanes -> 0 ----
  float F[5], Bb[5];
  {
    float xnext = __shfl_down(xw[0], 1, 32);
#pragma unroll
    for (int q = 0; q < 5; q++) {
      int j = lane * 5 + q;
      F[q]  = (j <= 158) ? ((q < 4) ? xw[q + 1] : xnext) : 0.f;
      Bb[q] = (j <= 158) ? xw[q] : 0.f;
    }
  }
  float den;
  {
    float d = 0.f;
#pragma unroll
    for (int q = 0; q < 5; q++) d += F[q] * F[q] + Bb[q] * Bb[q];
    den = wred32(d);
  }

  float a[ORDER + 1];
#pragma unroll
  for (int m = 0; m <= ORDER; m++) a[m] = (m == 0) ? 1.f : 0.f;

  int n = FRAME - 1;    // active window length (159, shrinks each iter)
#pragma unroll
  for (int i = 0; i < ORDER; i++) {
    float num = 0.f;
#pragma unroll
    for (int q = 0; q < 5; q++) num += F[q] * Bb[q];
    num = wred32(num);
    float k = -2.0f * num / den;

    // a[1..i+1] += k * a[i..0]  (uniform across lanes; fully unrolled)
    float atmp[ORDER + 1];
#pragma unroll
    for (int m = 0; m <= ORDER; m++) atmp[m] = a[m];
#pragma unroll
    for (int m = 1; m <= ORDER; m++)
      if (m <= i + 1) a[m] = atmp[m] + k * atmp[i + 1 - m];

    float tf[5], tb[5];
#pragma unroll
    for (int q = 0; q < 5; q++) {
      tf[q] = F[q] + k * Bb[q];
      tb[q] = Bb[q] + k * F[q];
    }
    // den = (1-k^2)*den - bwd_n[n-1]^2 - fwd_n[0]^2
    float tf0 = __shfl(tf[0], 0, 32);
    int jl = n - 1;
    int qs = jl % 5;
    float tsel = (qs == 0) ? tb[0] : (qs == 1) ? tb[1] : (qs == 2) ? tb[2]
               : (qs == 3) ? tb[3] : tb[4];
    float tbl = __shfl(tsel, jl / 5, 32);
    den = (1.0f - k * k) * den - tbl * tbl - tf0 * tf0;

    // fwd = fwd_n[1:], bwd = bwd_n[:-1]; mask dropped tail to zero
    float tfn = __shfl_down(tf[0], 1, 32);
    int nn = n - 1;
#pragma unroll
    for (int q = 0; q < 5; q++) {
      int j = lane * 5 + q;
      F[q]  = (j < nn) ? ((q < 4) ? tf[q + 1] : tfn) : 0.f;
      Bb[q] = (j < nn) ? tb[q] : 0.f;
    }
    n = nn;
  }

  // ---- residual FIR: res[t] = sum_kk a[kk]*xw[t-kk], zero front pad ----
  if (lane < 12) sx[wv][lane] = 0.f;
#pragma unroll
  for (int q = 0; q < 5; q++) sx[wv][12 + lane * 5 + q] = xw[q];
  asm volatile("" ::: "memory");   // wave-private LDS row: HW keeps DS ops in order

  float lsum = 0.f, lsq = 0.f;
#pragma unroll
  for (int q = 0; q < 5; q++) {
    int t = lane * 5 + q;
    float r = 0.f;
#pragma unroll
    for (int kk = 0; kk <= ORDER; kk++) r += a[kk] * sx[wv][12 + t - kk];
    resid[(size_t)fr * FRAME + t] = r;
    lsum += r;
    lsq  += r * r;
  }
  lsum = wred32(lsum);
  lsq  = wred32(lsq);
  if (lane == 0) { fsum[fr] = lsum; fsq[fr] = lsq; }
}

// =====================================================================
// Kernel 2: deterministic per-batch std of residual -> denom = std+1e-6
// =====================================================================
__global__ void batch_std_kernel(const float* __restrict__ fsum,
                                 const float* __restrict__ fsq,
                                 float* __restrict__ denom, int B)
{
  int b = blockIdx.x * blockDim.x + threadIdx.x;
  if (b >= B) return;
  float s = 0.f, q = 0.f;
#pragma unroll
  for (int k = 0; k < KFR; k++) { s += fsum[b * KFR + k]; q += fsq[b * KFR + k]; }
  float m   = s / (float)TLEN;
  float var = q / (float)TLEN - m * m;
  denom[b] = sqrtf(fmaxf(var, 0.f)) + 1e-6f;
}

// =====================================================================
// Kernel 3: one wave32 per 16 consecutive (b,p) blocks.
//   - block stats (mu/std of r, s_rms, r_rms) via lane-pair reductions
//   - MLP 1->16->1 via V_WMMA_F32_16X16X4_F32 (1 + 4 chained WMMAs)
//   - elementwise output clip(pcm + alpha*chips) from cached registers
// =====================================================================
__global__ __launch_bounds__(32)
void blocks_mlp_out_kernel(const float* __restrict__ pcm,
                           const float* __restrict__ resid,
                           const float* __restrict__ denom,
                           const int*   __restrict__ bits,
                           const float* __restrict__ W1,
                           const float* __restrict__ b1,
                           const float* __restrict__ W2,
                           const float* __restrict__ b2,
                           float* __restrict__ out)
{
  const int lane = threadIdx.x & 31;
  const int g0   = blockIdx.x * 16;       // first global block id (b*64+p)
  const int b    = g0 >> 6;               // 16 blocks never straddle a batch
  const int p0   = g0 & 63;

  const int i    = lane >> 1;             // block 0..15 within this wave
  const int half = lane & 1;              // each block handled by a lane pair
  const int p    = p0 + i;
  const int t0   = p * LBLK + half * 19;  // L=38 = 2*19 samples per block

  const float dinv = 1.0f / denom[b];

  // ---- single pass loads cached in registers for reuse at the end ----
  float sv[19], rv[19];
  float s_r = 0.f, s_r2 = 0.f, s_s2 = 0.f, s_rr2 = 0.f;
#pragma unroll
  for (int m = 0; m < 19; m++) {
    int t = t0 + m;
    float s = 0.f, rr = 0.f;
    if (t < TLEN) {
      s  = pcm[(size_t)b * TLEN + t];
      rr = resid[(size_t)b * TLEN + t];
    }
    sv[m] = s; rv[m] = rr;
    float r = rr * dinv;
    s_r += r; s_r2 += r * r; s_s2 += s * s; s_rr2 += rr * rr;
  }
  s_r  += __shfl_xor(s_r, 1, 32);
  s_r2 += __shfl_xor(s_r2, 1, 32);
  s_s2 += __shfl_xor(s_s2, 1, 32);
  s_rr2+= __shfl_xor(s_rr2, 1, 32);

  const float invL = 1.0f / (float)LBLK;
  float mu    = s_r * invL;
  float var   = s_r2 * invL - mu * mu;
  float stdb  = sqrtf(fmaxf(var, 0.f)) + 1e-6f;
  float s_rms = sqrtf(s_s2 * invL + 1e-8f);
  float r_rms = sqrtf(s_rr2 * invL + 1e-8f);
  float a_k   = (s_rms / r_rms) * SSL;
  float feat  = logf(s_rms + 1e-6f);

  // ================= MLP via WMMA f32 16x16x4 =================
  // Stage 1: H(16x16) = A(16x4) x B(4x16); A rows = [feat_m, 1, 0, 0],
  //          B rows  = [W1; b1; 0; 0]  =>  H[m][n] = feat_m*W1[n]+b1[n]
  float featg = __shfl(feat, (lane & 15) * 2, 32);  // block (lane&15)'s feat
  v2f Am, Bm;
  Am.x = (lane < 16) ? featg : 0.f;                 // K=0 (lo half) / K=2 (hi)
  Am.y = (lane < 16) ? 1.f   : 0.f;                 // K=1 / K=3
  Bm.x = (lane < 16) ? W1[lane & 15] : 0.f;
  Bm.y = (lane < 16) ? b1[lane & 15] : 0.f;
  v8f H = {0.f, 0.f, 0.f, 0.f, 0.f, 0.f, 0.f, 0.f};
  H = __builtin_amdgcn_wmma_f32_16x16x4_f32(false, Am, false, Bm,
                                            (short)0, H, false, false);
#pragma unroll
  for (int g = 0; g < 8; g++) H[g] = fmaxf(H[g], 0.f);   // relu

  // Stage 2: logits(16) = H(16x16) x W2(16) + b2, as 4 chained K=4 WMMAs
  // with W2 broadcast across all 16 columns (every column = same logit).
  __shared__ float hbuf[256];
  __shared__ float gv[16];
  {
    int Mb = (lane >= 16) ? 8 : 0;
    int N  = lane & 15;
#pragma unroll
    for (int g = 0; g < 8; g++) hbuf[(g + Mb) * 16 + N] = H[g];
  }
  __syncthreads();

  float b2s = b2[0];
  v8f acc = {b2s, b2s, b2s, b2s, b2s, b2s, b2s, b2s};
#pragma unroll
  for (int cix = 0; cix < 4; cix++) {
    int kb = 4 * cix + ((lane >= 16) ? 2 : 0);
    int M  = lane & 15;
    v2f A2, B2;
    A2.x = hbuf[M * 16 + kb];
    A2.y = hbuf[M * 16 + kb + 1];
    B2.x = W2[kb];
    B2.y = W2[kb + 1];
    acc = __builtin_amdgcn_wmma_f32_16x16x4_f32(false, A2, false, B2,
                                                (short)0, acc, false, false);
  }
  // lane 0 holds rows 0..7, lane 16 holds rows 8..15 (any column is valid)
  if (lane == 0) {
#pragma unroll
    for (int g = 0; g < 8; g++) gv[g] = 1.f / (1.f + expf(-acc[g]));
  }
  if (lane == 16) {
#pragma unroll
    for (int g = 0; g < 8; g++) gv[8 + g] = 1.f / (1.f + expf(-acc[g]));
  }
  __syncthreads();

  // ================= elementwise output =================
  float alpha = a_k * gv[i];
  float sgn   = (float)(bits[(size_t)b * PCH + p] * 2 - 1);
  float istd  = 1.0f / stdb;
#pragma unroll
  for (int m = 0; m < 19; m++) {
    int t = t0 + m;
    if (t < TLEN) {
      float r    = rv[m] * dinv;
      float chip = (r - mu) * istd * sgn;
      float o    = sv[m] + alpha * chip;
      out[(size_t)b * TLEN + t] = fminf(fmaxf(o, -1.f), 1.f);
    }
  }
}

// =====================================================================
extern "C" void kernel_launch(void* const* d_in, const int* in_sizes, int n_in,
                              void* d_out, int out_size, void* d_ws, size_t ws_size,
                              hipStream_t stream)
{
  const int*   bits = (const int*)  d_in[0];
  const float* pcm  = (const float*)d_in[1];
  const float* W1   = (const float*)d_in[2];
  const float* b1   = (const float*)d_in[3];
  const float* W2   = (const float*)d_in[4];
  const float* b2   = (const float*)d_in[5];
  float* out = (float*)d_out;

  const int B       = in_sizes[0] / PCH;   // 4096
  const int nframes = B * KFR;             // 61440

  // workspace: resid[B*T] | fsum[nframes] | fsq[nframes] | denom[B]
  // total ~= 39.3 MB + 0.5 MB  (everything fully overwritten each call)
  float* resid = (float*)d_ws;
  float* fsum  = resid + (size_t)B * TLEN;
  float* fsq   = fsum + nframes;
  float* denom = fsq + nframes;

  burg_resid_kernel<<<(nframes + 7) / 8, 256, 0, stream>>>(
      pcm, resid, fsum, fsq, nframes);
  batch_std_kernel<<<(B + 255) / 256, 256, 0, stream>>>(fsum, fsq, denom, B);
  blocks_mlp_out_kernel<<<(B * PCH) / 16, 32, 0, stream>>>(
      pcm, resid, denom, bits, W1, b1, W2, b2, out);
}